// GCBFSafetyLayer_4097398800551
// MI455X (gfx1250) — compile-verified
//
#include <hip/hip_runtime.h>
#include <math.h>

// CDNA5 (gfx1250, wave32) vector types for WMMA operands.
typedef float v2f __attribute__((ext_vector_type(2)));
typedef float v8f __attribute__((ext_vector_type(8)));

#define NAGENT 1024
#define NOBS   256
#define NCOLS  (NAGENT + NOBS)   // 1280
// Block tile: 32 rows x 64 cols, 8 waves, one 16x16 WMMA tile per wave.
#define ROWS_PER_BLK 32
#define COLS_PER_BLK 64

// rhs[b,n,c] = s_n . q_c + a_n + b_c      (K=4 GEMM + row/col bias)
//   s_n = 2*(p_n+v_n), a_n = -|p_n|^2 - 2 p_n.v_n, b_c = thr_c - |q_c|^2
// diag(n==c) -> -inf.
__global__ __launch_bounds__(256) void gcbf_rhs_wmma(
    const float* __restrict__ pos,   // [B, 1024, 3]
    const float* __restrict__ vel,   // [B, 1024, 3]
    const float* __restrict__ obs,   // [B, 256, 4]
    float* __restrict__ out)         // [B, 1024, 1280]
{
    __shared__ float4 sA[ROWS_PER_BLK];   // [sx, sy, sz, a_n]  (A rows, K=4)
    __shared__ float4 sB[COLS_PER_BLK];   // [qx, qy, qz, 1.0]  (B cols, K=4)
    __shared__ float  sBias[COLS_PER_BLK];

    const int tid     = threadIdx.x;
    const int batch   = blockIdx.z;
    const int colBase = blockIdx.x * COLS_PER_BLK;
    const int rowBase = blockIdx.y * ROWS_PER_BLK;

    const float* P = pos + (size_t)batch * NAGENT * 3;
    const float* V = vel + (size_t)batch * NAGENT * 3;
    const float* O = obs + (size_t)batch * NOBS * 4;

    // ---- Stage the rank-4 factors into LDS (tiny: ~1.75 KB) ----
    if (tid < ROWS_PER_BLK) {
        const int r = rowBase + tid;
        const float px = P[r * 3 + 0], py = P[r * 3 + 1], pz = P[r * 3 + 2];
        const float vx = V[r * 3 + 0], vy = V[r * 3 + 1], vz = V[r * 3 + 2];
        const float a = -(px * px + py * py + pz * pz)
                        - 2.0f * (px * vx + py * vy + pz * vz);
        sA[tid] = make_float4(2.0f * (px + vx), 2.0f * (py + vy),
                              2.0f * (pz + vz), a);
    } else if (tid >= 64 && tid < 64 + COLS_PER_BLK) {
        const int ci = tid - 64;
        const int c  = colBase + ci;
        float qx, qy, qz, thr;
        if (c < NAGENT) {
            qx = P[c * 3 + 0]; qy = P[c * 3 + 1]; qz = P[c * 3 + 2];
            thr = 0.0225f;                       // (2*0.05 + 0.05)^2
        } else {
            const float4 o = ((const float4*)O)[c - NAGENT];
            qx = o.x; qy = o.y; qz = o.z;
            const float rr = o.w + 0.1f;         // r + AGENT_R + MARGIN
            thr = rr * rr;
        }
        sB[ci]    = make_float4(qx, qy, qz, 1.0f);
        sBias[ci] = thr - (qx * qx + qy * qy + qz * qz);
    }
    __syncthreads();

    // ---- One 16x16x4 f32 WMMA per wave ----
    const int wave    = tid >> 5;
    const int lane    = tid & 31;
    const int rowTile = wave >> 2;      // 0..1
    const int colTile = wave & 3;       // 0..3
    const int half    = lane >> 4;      // 0: K=0,1 / rows+0 ; 1: K=2,3 / rows+8
    const int l15     = lane & 15;

    // A fragment: lane holds A[M=l15][K = 2*half .. 2*half+1]  (ds_load_b64)
    const float* ap = (const float*)&sA[rowTile * 16 + l15];
    v2f aF; aF.x = ap[half * 2 + 0]; aF.y = ap[half * 2 + 1];
    // B fragment: lane holds B[K = 2*half .. 2*half+1][N=l15]
    const float* bp = (const float*)&sB[colTile * 16 + l15];
    v2f bF; bF.x = bp[half * 2 + 0]; bF.y = bp[half * 2 + 1];

    const float bias = sBias[colTile * 16 + l15];

    v8f acc = {};
    // (neg_a, A, neg_b, B, c_mod, C, reuse_a, reuse_b)
    acc = __builtin_amdgcn_wmma_f32_16x16x4_f32(
        false, aF, false, bF, (short)0, acc, false, false);

    // D layout: VGPR r -> row rowBase+rowTile*16 + r + 8*half, col = l15 lane.
    const int col  = colBase + colTile * 16 + l15;
    const int row0 = rowBase + rowTile * 16 + half * 8;
    float* outp = out + ((size_t)batch * NAGENT + row0) * NCOLS + col;

#pragma unroll
    for (int r = 0; r < 8; ++r) {
        float vvl = acc[r] + bias;
        if (row0 + r == col) vvl = -__builtin_inff();   // agent-self -> -inf
        outp[(size_t)r * NCOLS] = vvl;                  // 64B-contig per half
    }
}

extern "C" void kernel_launch(void* const* d_in, const int* in_sizes, int n_in,
                              void* d_out, int out_size, void* d_ws, size_t ws_size,
                              hipStream_t stream) {
    const float* pos = (const float*)d_in[0];  // [B,1024,3] f32
    const float* vel = (const float*)d_in[1];  // [B,1024,3] f32
    const float* obs = (const float*)d_in[2];  // [B,256,4]  f32
    float* out = (float*)d_out;                // [B,1024,1280] f32

    const int B = in_sizes[0] / (NAGENT * 3);  // = 8
    dim3 block(256, 1, 1);
    dim3 grid(NCOLS / COLS_PER_BLK,            // 20 col-blocks
              NAGENT / ROWS_PER_BLK,           // 32 row-blocks
              B);                              // 8 batches
    gcbf_rhs_wmma<<<grid, block, 0, stream>>>(pos, vel, obs, out);
}